// UMLSGNN_36206574305711
// MI455X (gfx1250) — compile-verified
//
#include <hip/hip_runtime.h>
#include <cstddef>

#define N_CUI  100000
#define N_CODE 50000
#define DIM    128
#define E_CC   1600000
#define E_KC   800000
#define E_CK   800000

typedef __attribute__((ext_vector_type(2))) float v2f;
typedef __attribute__((ext_vector_type(8))) float v8f;

// ---------------- degree kernels ----------------
__global__ void umls_count_deg(const int* __restrict__ ei, int E,
                               float* __restrict__ cnt) {
  int i = blockIdx.x * blockDim.x + threadIdx.x;
  if (i < E) unsafeAtomicAdd(&cnt[ei[E + i]], 1.0f);  // row 1 = dst
}

__global__ void umls_invert_deg(float* __restrict__ cnt, int n) {
  int i = blockIdx.x * blockDim.x + threadIdx.x;
  if (i < n) cnt[i] = 1.0f / fmaxf(cnt[i], 1.0f);
}

// ---------------- edge gather/scatter-add (one wave per edge) ----------------
__global__ __launch_bounds__(256) void umls_scatter_accum(
    const float* __restrict__ xsrc, const int* __restrict__ ei, int E,
    float* __restrict__ agg) {
  int gid  = blockIdx.x * blockDim.x + threadIdx.x;
  int edge = gid >> 5;
  int lane = gid & 31;
  if (edge >= E) return;
  int src = ei[edge];        // row 0 = src
  int dst = ei[E + edge];    // row 1 = dst
  float4 v = *(reinterpret_cast<const float4*>(xsrc + (size_t)src * DIM) + lane);
  float* dp = agg + (size_t)dst * DIM + lane * 4;
  unsafeAtomicAdd(dp + 0, v.x);
  unsafeAtomicAdd(dp + 1, v.y);
  unsafeAtomicAdd(dp + 2, v.z);
  unsafeAtomicAdd(dp + 3, v.w);
}

// ---------------- fused SAGE GEMM via V_WMMA_F32_16X16X4_F32 ----------------
// Pure load->wmma inner loop: acc = A @ W^T over K=128 (steps of 4).
// Per-row mean scaling (1/deg) is applied in the epilogue instead of on the
// A fragments, so no VALU sits in the WMMA dependency chain.
__device__ __forceinline__ v8f umls_wmma_k128(const float* __restrict__ A,
                                              const float* __restrict__ W,
                                              int m, int n, int kh) {
  const float* ar = A + (size_t)m * DIM + kh;
  const float* wr = W + (size_t)n * DIM + kh;
  v8f acc = {};
#pragma unroll
  for (int k0 = 0; k0 < DIM; k0 += 4) {
    // A 16x4 frag: lanes 0-15 carry K=k0,k0+1; lanes 16-31 carry K=k0+2,k0+3
    v2f a; a.x = ar[k0]; a.y = ar[k0 + 1];
    // B 4x16 frag: same K split, lane%16 selects column N (= output feature)
    v2f b; b.x = wr[k0]; b.y = wr[k0 + 1];
    acc = __builtin_amdgcn_wmma_f32_16x16x4_f32(false, a, false, b,
                                                (short)0, acc, false, false);
  }
  return acc;
}

// mode: 0 = store, 1 = relu+store, 2 = elementwise max into Out
__global__ __launch_bounds__(256) void umls_sage_gemm(
    const float* __restrict__ A0, const float* __restrict__ inv0, const float* __restrict__ W0,
    const float* __restrict__ X0, const float* __restrict__ Wr0,
    const float* __restrict__ A1, const float* __restrict__ inv1, const float* __restrict__ W1,
    const float* __restrict__ Wr1,
    const float* __restrict__ bias0, const float* __restrict__ bias1,
    float* __restrict__ Out, int mode) {
  int lane = threadIdx.x & 31;
  int wave = threadIdx.x >> 5;       // 8 waves/block, one 16-wide N tile each
  int mrow = blockIdx.x * 16;        // 16 nodes per block
  int ncol = wave * 16;
  int t    = lane & 15;
  int kh   = (lane >> 4) << 1;       // 0 for lanes 0-15, 2 for lanes 16-31
  int m    = mrow + t;               // A row this lane loads
  int n    = ncol + t;               // W row (output feature) this lane loads

  // one accumulator per product so 1/deg scaling can move to the epilogue
  v8f acc0 = umls_wmma_k128(A0, W0, m, n, kh);   // lin_l(sum_cc)  (scale later)
  v8f acc1 = umls_wmma_k128(X0, Wr0, m, n, kh);  // lin_r(x_dst)
  v8f acc2 = {};
  v8f acc3 = {};
  if (A1 != nullptr) {                           // second conv (hetero sum)
    acc2 = umls_wmma_k128(A1, W1, m, n, kh);     // lin_l(sum_kc)  (scale later)
    acc3 = umls_wmma_k128(X0, Wr1, m, n, kh);    // lin_r(x_dst)
  }

  float bv = bias0[n] + ((bias1 != nullptr) ? bias1[n] : 0.0f);
  int mbase = mrow + ((lane >> 4) << 3);  // C/D layout: +8 rows for lanes>=16
#pragma unroll
  for (int r = 0; r < 8; ++r) {
    int row = mbase + r;
    float val = inv0[row] * acc0[r] + acc1[r] + bv;
    if (A1 != nullptr) val += inv1[row] * acc2[r] + acc3[r];
    if (mode == 1) val = fmaxf(val, 0.0f);
    size_t idx = (size_t)row * DIM + (size_t)n;
    if (mode == 2) Out[idx] = fmaxf(Out[idx], val);
    else           Out[idx] = val;
  }
}

extern "C" void kernel_launch(void* const* d_in, const int* in_sizes, int n_in,
                              void* d_out, int out_size, void* d_ws, size_t ws_size,
                              hipStream_t stream) {
  const float* x_cui  = (const float*)d_in[0];
  const float* x_code = (const float*)d_in[1];
  const int*   ei_cc  = (const int*)d_in[2];
  const int*   ei_kc  = (const int*)d_in[3];
  const int*   ei_ck  = (const int*)d_in[4];
  const float* Wl     = (const float*)d_in[5];  // [F,L,3,128,128]
  const float* Wr     = (const float*)d_in[6];
  const float* bi     = (const float*)d_in[7];  // [F,L,3,128]

  float* out_cui  = (float*)d_out;
  float* out_code = out_cui + (size_t)N_CUI * DIM;

  // workspace carve (contiguous so phases can memset in one shot)
  float* p = (float*)d_ws;
  float* inv_cc = p; p += N_CUI;
  float* inv_kc = p; p += N_CUI;
  float* inv_ck = p; p += N_CODE;
  float* agg_cc = p; p += (size_t)N_CUI * DIM;
  float* agg_kc = p; p += (size_t)N_CUI * DIM;
  float* agg_ck = p; p += (size_t)N_CODE * DIM;
  float* h1_cui = p; p += (size_t)N_CUI * DIM;
  float* h1_code = p; p += (size_t)N_CODE * DIM;

  const int T = 256;

  // ---- degrees: edge structure is fixed across filters/layers, do once ----
  hipMemsetAsync(inv_cc, 0, (size_t)(2 * N_CUI + N_CODE) * sizeof(float), stream);
  umls_count_deg<<<(E_CC + T - 1) / T, T, 0, stream>>>(ei_cc, E_CC, inv_cc);
  umls_count_deg<<<(E_KC + T - 1) / T, T, 0, stream>>>(ei_kc, E_KC, inv_kc);
  umls_count_deg<<<(E_CK + T - 1) / T, T, 0, stream>>>(ei_ck, E_CK, inv_ck);
  umls_invert_deg<<<(2 * N_CUI + N_CODE + T - 1) / T, T, 0, stream>>>(
      inv_cc, 2 * N_CUI + N_CODE);

  const size_t WSZ = (size_t)DIM * DIM;
  for (int f = 0; f < 2; ++f) {
    const float* h_cui  = x_cui;
    const float* h_code = x_code;
    for (int l = 0; l < 2; ++l) {
      // zero all three agg buffers (contiguous)
      hipMemsetAsync(agg_cc, 0,
                     (size_t)(2 * N_CUI + N_CODE) * DIM * sizeof(float), stream);
      umls_scatter_accum<<<(size_t)E_CC * 32 / T, T, 0, stream>>>(h_cui,  ei_cc, E_CC, agg_cc);
      umls_scatter_accum<<<(size_t)E_KC * 32 / T, T, 0, stream>>>(h_code, ei_kc, E_KC, agg_kc);
      umls_scatter_accum<<<(size_t)E_CK * 32 / T, T, 0, stream>>>(h_cui,  ei_ck, E_CK, agg_ck);

      size_t base = ((size_t)(f * 2 + l)) * 3;
      const float* Wl0 = Wl + (base + 0) * WSZ;
      const float* Wl1 = Wl + (base + 1) * WSZ;
      const float* Wl2 = Wl + (base + 2) * WSZ;
      const float* Wr0 = Wr + (base + 0) * WSZ;
      const float* Wr1 = Wr + (base + 1) * WSZ;
      const float* Wr2 = Wr + (base + 2) * WSZ;
      const float* b0  = bi + (base + 0) * DIM;
      const float* b1  = bi + (base + 1) * DIM;
      const float* b2  = bi + (base + 2) * DIM;

      int mode = (l == 0) ? 1 : ((f == 0) ? 0 : 2);  // relu | store | max
      float* o_cui  = (l == 0) ? h1_cui  : out_cui;
      float* o_code = (l == 0) ? h1_code : out_code;

      // cui destination: sum of cc-conv and kc-conv (4 fused WMMA products)
      umls_sage_gemm<<<N_CUI / 16, 256, 0, stream>>>(
          agg_cc, inv_cc, Wl0, h_cui, Wr0,
          agg_kc, inv_kc, Wl1, Wr1,
          b0, b1, o_cui, mode);
      // code destination: single ck-conv (2 fused WMMA products)
      umls_sage_gemm<<<N_CODE / 16, 256, 0, stream>>>(
          agg_ck, inv_ck, Wl2, h_code, Wr2,
          nullptr, nullptr, nullptr, nullptr,
          b2, nullptr, o_code, mode);

      h_cui = h1_cui;
      h_code = h1_code;
    }
  }
}